// LightGCN_6012954214604
// MI455X (gfx1250) — compile-verified
//
#include <hip/hip_runtime.h>
#include <hip/hip_bf16.h>
#include <cstdint>

typedef __attribute__((ext_vector_type(2))) float v2f;
typedef __attribute__((ext_vector_type(8))) float v8f;

#define TPB 256
#define EPB 16   // edges per block in scatter (256 threads / 16 lanes per edge)

// ---------------- utility kernels ----------------

__global__ void k_zero_u32(uint32_t* __restrict__ p, int n) {
    int i = blockIdx.x * blockDim.x + threadIdx.x;
    if (i < n) p[i] = 0u;
}

__global__ void k_zero_f4(float4* __restrict__ p, int n4) {
    int i = blockIdx.x * blockDim.x + threadIdx.x;
    if (i < n4) p[i] = make_float4(0.f, 0.f, 0.f, 0.f);
}

__global__ void k_copy_f4(float4* __restrict__ dst, const float4* __restrict__ src, int n4) {
    int i = blockIdx.x * blockDim.x + threadIdx.x;
    if (i < n4) dst[i] = src[i];
}

__global__ void k_accum_f4(float4* __restrict__ dst, const float4* __restrict__ src, int n4) {
    int i = blockIdx.x * blockDim.x + threadIdx.x;
    if (i < n4) {
        float4 d = dst[i];
        float4 s = src[i];
        d.x += s.x; d.y += s.y; d.z += s.z; d.w += s.w;
        dst[i] = d;
    }
}

// scalar 0.25 scale fallback (only used if N % 16 != 0)
__global__ void k_scale_f4(float4* __restrict__ p, int n4) {
    int i = blockIdx.x * blockDim.x + threadIdx.x;
    if (i < n4) {
        float4 v = p[i];
        v.x *= 0.25f; v.y *= 0.25f; v.z *= 0.25f; v.w *= 0.25f;
        p[i] = v;
    }
}

// ---------------- graph preprocessing ----------------

// deg[row[e]] += 1; deg[col[e]] += 1  (exact integer counts)
// Edge streams are read-once: NT hint keeps them out of the L2 hot set.
__global__ void k_degree(const int* __restrict__ row, const int* __restrict__ col,
                         uint32_t* __restrict__ deg, int E) {
    int e = blockIdx.x * blockDim.x + threadIdx.x;
    if (e < E) {
        int r = __builtin_nontemporal_load(&row[e]);
        int c = __builtin_nontemporal_load(&col[e]);
        atomicAdd(&deg[r], 1u);
        atomicAdd(&deg[c], 1u);
    }
}

// dis[n] = (max(deg,1))^-0.5 ; IEEE sqrt + divide for accuracy (N is tiny)
__global__ void k_dis(const uint32_t* __restrict__ deg, float* __restrict__ dis, int N) {
    int i = blockIdx.x * blockDim.x + threadIdx.x;
    if (i < N) {
        uint32_t d = deg[i];
        if (d < 1u) d = 1u;
        dis[i] = 1.0f / sqrtf((float)d);
    }
}

__global__ void k_norm(const int* __restrict__ row, const int* __restrict__ col,
                       const float* __restrict__ dis, float* __restrict__ norm, int E) {
    int e = blockIdx.x * blockDim.x + threadIdx.x;
    if (e < E) {
        int r = __builtin_nontemporal_load(&row[e]);
        int c = __builtin_nontemporal_load(&col[e]);
        norm[e] = dis[r] * dis[c];   // dis is 1.2MB -> L2/L0 resident
    }
}

// ---------------- propagation: gather-scale-scatter ----------------
// 16 lanes per edge, each lane owns a float4 (16*4 = 64 = embedding dim).
// Per block: 16 edges; their (row, col, norm) scalars are staged into LDS via
// the gfx1250 async-DMA path (global_load_async_to_lds_b32, ASYNCcnt), then
// broadcast-read from LDS by all 256 threads -> 16x fewer stream requests.
// Gather: one contiguous 256B row read (RT, L2-resident). Scatter: 4
// no-return relaxed/agent f32 atomics per lane -> hardware
// global_atomic_add_f32 RMW in L2 (src+dst tables = 153.6MB < 192MB L2).
__global__ void k_scatter(const int* __restrict__ row, const int* __restrict__ col,
                          const float* __restrict__ norm,
                          const float* __restrict__ src, float* __restrict__ dst, int E) {
    __shared__ int   s_r[EPB];
    __shared__ int   s_c[EPB];
    __shared__ float s_n[EPB];

    const int tid = threadIdx.x;
    const int e0  = blockIdx.x * EPB;

    // --- async stage 16 edges' scalars into LDS ---
    if (tid < 3 * EPB) {
        int i     = tid & (EPB - 1);
        int which = tid >> 4;            // 0: row, 1: col, 2: norm
        int e     = e0 + i;
        if (e < E) {
            uint64_t gaddr;
            uint32_t laddr;
            if (which == 0)      { gaddr = (uint64_t)(uintptr_t)&row[e];  laddr = (uint32_t)(uintptr_t)&s_r[i]; }
            else if (which == 1) { gaddr = (uint64_t)(uintptr_t)&col[e];  laddr = (uint32_t)(uintptr_t)&s_c[i]; }
            else                 { gaddr = (uint64_t)(uintptr_t)&norm[e]; laddr = (uint32_t)(uintptr_t)&s_n[i]; }
            // GV mode: per-lane 64-bit global address, LDS byte address in VDST
            asm volatile("global_load_async_to_lds_b32 %0, %1, off"
                         :: "v"(laddr), "v"(gaddr)
                         : "memory");
        }
    }
    asm volatile("s_wait_asynccnt 0x0" ::: "memory");
    __syncthreads();

    // --- process ---
    const int eg   = tid >> 4;           // edge slot within block
    const int lane = tid & 15;           // float4 slot within the 64-wide row
    const int e    = e0 + eg;
    if (e < E) {
        int   r = s_r[eg];
        int   c = s_c[eg];
        float s = s_n[eg];
        const float4* sp = reinterpret_cast<const float4*>(src + (size_t)c * 64) + lane;
        float4 v = *sp;
        float* dp = dst + (size_t)r * 64 + (size_t)lane * 4;
        __hip_atomic_fetch_add(dp + 0, v.x * s, __ATOMIC_RELAXED, __HIP_MEMORY_SCOPE_AGENT);
        __hip_atomic_fetch_add(dp + 1, v.y * s, __ATOMIC_RELAXED, __HIP_MEMORY_SCOPE_AGENT);
        __hip_atomic_fetch_add(dp + 2, v.z * s, __ATOMIC_RELAXED, __HIP_MEMORY_SCOPE_AGENT);
        __hip_atomic_fetch_add(dp + 3, v.w * s, __ATOMIC_RELAXED, __HIP_MEMORY_SCOPE_AGENT);
    }
}

// ---------------- final mean via WMMA f32 16x16x4 ----------------
// out *= 0.25 on 16x16 tiles. D = sum_k A_k x B_k with B_k = 0.25 * one-hot
// identity column-block; exactly one nonzero product per output element and a
// power-of-two scale -> bit-exact 0.25*x. One wave per tile; guard is
// wave-uniform so EXEC is all ones at every v_wmma.
__global__ void k_scale_wmma(float* __restrict__ out, int totalTiles) {
    int wave = blockIdx.x * (blockDim.x >> 5) + (threadIdx.x >> 5);
    if (wave >= totalTiles) return;           // uniform within the wave
    int lane = threadIdx.x & 31;
    int tr = wave >> 2;                       // 16-row block
    int tc = wave & 3;                        // 16-col block within d=64
    float* base = out + (size_t)tr * 16 * 64 + (size_t)tc * 16;

    int n    = lane & 15;                     // column for B/C/D, row for A
    int half = lane >> 4;
    int kk   = half * 2;                      // A holds K pair {0,1} or {2,3}

    v8f acc = {};
#pragma unroll
    for (int k = 0; k < 4; ++k) {
        const float* arow = base + (size_t)n * 64 + (size_t)k * 4 + kk;
        v2f a; a.x = arow[0]; a.y = arow[1];
        v2f b;
        b.x = (n == k * 4 + kk + 0) ? 0.25f : 0.0f;
        b.y = (n == k * 4 + kk + 1) ? 0.25f : 0.0f;
        acc = __builtin_amdgcn_wmma_f32_16x16x4_f32(
            /*neg_a=*/false, a, /*neg_b=*/false, b,
            /*c_mod=*/(short)0, acc, /*reuse_a=*/false, /*reuse_b=*/false);
    }
#pragma unroll
    for (int j = 0; j < 8; ++j)
        base[(size_t)(8 * half + j) * 64 + n] = acc[j];
}

// ---------------- host launcher ----------------

static inline int cdiv(long long a, int b) { return (int)((a + b - 1) / b); }

extern "C" void kernel_launch(void* const* d_in, const int* in_sizes, int n_in,
                              void* d_out, int out_size, void* d_ws, size_t ws_size,
                              hipStream_t stream) {
    const int*   eidx = (const int*)d_in[0];
    const float* emb  = (const float*)d_in[1];
    const int E  = in_sizes[0] / 2;
    const int Nd = in_sizes[1];          // N * 64
    const int N  = Nd / 64;
    const int n4 = Nd / 4;

    const int* row = eidx;
    const int* col = eidx + E;

    // Workspace carve-out (256B aligned slices)
    char*  ws  = (char*)d_ws;
    size_t off = 0;
    auto carve = [&](size_t bytes) {
        void* p = ws + off;
        off += (bytes + 255) & ~(size_t)255;
        return p;
    };
    uint32_t* deg  = (uint32_t*)carve((size_t)N * 4);
    float*    dis  = (float*)carve((size_t)N * 4);
    float*    norm = (float*)carve((size_t)E * 4);
    float*    embA = (float*)carve((size_t)Nd * 4);
    float*    embB = (float*)carve((size_t)Nd * 4);
    (void)ws_size; (void)n_in; (void)out_size;

    float* out = (float*)d_out;

    // --- degree / normalization coefficients ---
    k_zero_u32<<<cdiv(N, TPB), TPB, 0, stream>>>(deg, N);
    k_degree<<<cdiv(E, TPB), TPB, 0, stream>>>(row, col, deg, E);
    k_dis<<<cdiv(N, TPB), TPB, 0, stream>>>(deg, dis, N);
    k_norm<<<cdiv(E, TPB), TPB, 0, stream>>>(row, col, dis, norm, E);

    // --- layer_sum = embedding (layer 0) ---
    k_copy_f4<<<cdiv(n4, TPB), TPB, 0, stream>>>((float4*)out, (const float4*)emb, n4);

    // --- 3 propagation layers, ping-pong A/B ---
    const float* src = emb;
    float* dst = embA;
    for (int l = 0; l < 3; ++l) {
        k_zero_f4<<<cdiv(n4, TPB), TPB, 0, stream>>>((float4*)dst, n4);
        k_scatter<<<cdiv(E, EPB), TPB, 0, stream>>>(row, col, norm, src, dst, E);
        k_accum_f4<<<cdiv(n4, TPB), TPB, 0, stream>>>((float4*)out, (const float4*)dst, n4);
        src = dst;
        dst = (dst == embA) ? embB : embA;
    }

    // --- final mean: out *= 0.25 via WMMA tiles (N % 16 == 0, d == 64) ---
    if ((N & 15) == 0) {
        int totalTiles  = (N / 16) * 4;
        int wavesPerBlk = TPB / 32;
        k_scale_wmma<<<cdiv(totalTiles, wavesPerBlk), TPB, 0, stream>>>(out, totalTiles);
    } else {
        k_scale_f4<<<cdiv(n4, TPB), TPB, 0, stream>>>((float4*)out, n4);
    }
}